// MPULoss_V2_23218593202209
// MI455X (gfx1250) — compile-verified
//
#include <hip/hip_runtime.h>
#include <hip/hip_bf16.h>
#include <math.h>

#define NCLASS          1000
#define NROWS           65536
#define ROWS_PER_TILE   8
#define TILES_PER_BLOCK 8
#define NBLOCKS         1024          // NROWS / (ROWS_PER_TILE * TILES_PER_BLOCK)
#define F4_PER_ROW      250           // 1000 / 4

typedef unsigned int u32v4 __attribute__((ext_vector_type(4)));
typedef int          i32v4 __attribute__((ext_vector_type(4)));
typedef int          i32v8 __attribute__((ext_vector_type(8)));

// Issue one TDM load of `nrows` rows x NCLASS f32 from global into LDS.
// D# packing per CDNA5 ISA sec 8.3/8.4 (2D tile, no pad, no iterate, no cluster).
__device__ __forceinline__ void tdm_load_rows(const float* gsrc,
                                              unsigned lds_off_bytes,
                                              int nrows)
{
    unsigned long long ga = (unsigned long long)(uintptr_t)gsrc;

    u32v4 g0;
    g0[0] = 1u;                                    // count=1 (valid user D#)
    g0[1] = lds_off_bytes;                         // lds_addr (bytes)
    g0[2] = (unsigned)(ga & 0xFFFFFFFFu);          // global_addr[31:0]
    g0[3] = (unsigned)((ga >> 32) & 0x01FFFFFFu)   // global_addr[56:32]
          | (2u << 30);                            // type = 2 ("image")

    i32v8 g1;
    g1[0] = (int)(2u << 16);                       // data_size=4B; wg_mask=0
    g1[1] = (int)((unsigned)NCLASS << 16);         // tensor_dim0[15:0] @ bits 63:48
    g1[2] = (int)((unsigned)nrows  << 16);         // tensor_dim0[31:16]=0 | tensor_dim1[15:0]
    g1[3] = (int)((unsigned)NCLASS << 16);         // tensor_dim1[31:16]=0 | tile_dim0
    g1[4] = nrows;                                 // tile_dim1 (tile_dim2 = 0)
    g1[5] = NCLASS;                                // tensor_dim0_stride[31:0]
    g1[6] = 0;                                     // stride0[47:32] | stride1[15:0]
    g1[7] = 0;                                     // stride1[47:16]

    i32v4 z4 = {0, 0, 0, 0};
#if __has_include(<hip/amd_detail/amd_gfx1250_TDM.h>)
    i32v8 z8 = {0, 0, 0, 0, 0, 0, 0, 0};
    __builtin_amdgcn_tensor_load_to_lds(g0, g1, z4, z4, z8, 0);
#else
    __builtin_amdgcn_tensor_load_to_lds(g0, g1, z4, z4, 0);
#endif
}

__global__ __launch_bounds__(256) void mpu_main(const float* __restrict__ outputs,
                                                const int*   __restrict__ labels,
                                                const float* __restrict__ priorlist,
                                                float*       __restrict__ partials)
{
    __shared__ __align__(16) float lbuf[2][ROWS_PER_TILE][NCLASS];  // 64 000 B, double-buffered
    __shared__ float red[ROWS_PER_TILE][3];

    const int tid  = threadIdx.x;
    const int wave = tid >> 5;
    const int lane = tid & 31;
    const int row0 = blockIdx.x * (ROWS_PER_TILE * TILES_PER_BLOCK);

    float pu2_acc = 0.0f, ce_acc = 0.0f, cnt_acc = 0.0f;

    // Prologue: wave 0 issues the DMA for tile 0.
    if (tid < 32) {
        tdm_load_rows(outputs + (size_t)row0 * NCLASS,
                      (unsigned)(uintptr_t)&lbuf[0][0][0], ROWS_PER_TILE);
    }

    for (int t = 0; t < TILES_PER_BLOCK; ++t) {
        const int cur = t & 1;

        if (tid < 32) {
            __builtin_amdgcn_s_wait_tensorcnt(0);   // tile t landed in LDS
        }
        __syncthreads();                             // publish to all waves; also
                                                     // guarantees buf[cur^1] reads done
        if ((tid < 32) && (t + 1) < TILES_PER_BLOCK) {
            tdm_load_rows(outputs + (size_t)(row0 + (t + 1) * ROWS_PER_TILE) * NCLASS,
                          (unsigned)(uintptr_t)&lbuf[cur ^ 1][0][0], ROWS_PER_TILE);
        }

        // ---- each wave reduces one row of the tile (register-resident) ----
        const float4* rowp = (const float4*)&lbuf[cur][wave][0];
        float4 v[8];
        float  m = -INFINITY;
        #pragma unroll
        for (int it = 0; it < 8; ++it) {
            const int idx = it * 32 + lane;
            if (idx < F4_PER_ROW) {
                v[it] = rowp[idx];                   // ds_load_b128, conflict-free
                m = fmaxf(m, fmaxf(fmaxf(v[it].x, v[it].y), fmaxf(v[it].z, v[it].w)));
            }
        }
        #pragma unroll
        for (int s = 16; s >= 1; s >>= 1)
            m = fmaxf(m, __shfl_xor(m, s, 32));

        float sum = 0.0f;
        #pragma unroll
        for (int it = 0; it < 8; ++it) {
            const int idx = it * 32 + lane;
            if (idx < F4_PER_ROW) {
                sum += __expf(v[it].x - m) + __expf(v[it].y - m)
                     + __expf(v[it].z - m) + __expf(v[it].w - m);
            }
        }
        #pragma unroll
        for (int s = 16; s >= 1; s >>= 1)
            sum += __shfl_xor(sum, s, 32);

        if (lane == 0) {
            const int r   = row0 + t * ROWS_PER_TILE + wave;
            const int lab = labels[r];
            if (lab < NCLASS) {                      // positive sample
                const float xl = lbuf[cur][wave][lab];
                const float lp = xl - m - __logf(sum);   // log-softmax at label
                const float p  = __expf(lp);             // softmax prob
                pu2_acc += -__logf(1.01f - p) * priorlist[lab];
                ce_acc  += -lp;
                cnt_acc += 1.0f;
            }
        }
    }

    if (lane == 0) { red[wave][0] = pu2_acc; red[wave][1] = ce_acc; red[wave][2] = cnt_acc; }
    __syncthreads();
    if (tid == 0) {
        float a = 0.0f, b = 0.0f, c = 0.0f;
        for (int w = 0; w < ROWS_PER_TILE; ++w) { a += red[w][0]; b += red[w][1]; c += red[w][2]; }
        partials[blockIdx.x * 4 + 0] = a;
        partials[blockIdx.x * 4 + 1] = b;
        partials[blockIdx.x * 4 + 2] = c;
    }
}

__global__ __launch_bounds__(256) void mpu_finalize(const float* __restrict__ partials,
                                                    float* __restrict__ out)
{
    __shared__ float sred[3][8];
    const int tid = threadIdx.x, lane = tid & 31, wave = tid >> 5;

    float a = 0.0f, b = 0.0f, c = 0.0f;
    for (int i = tid; i < NBLOCKS; i += 256) {       // fixed order -> deterministic
        a += partials[i * 4 + 0];
        b += partials[i * 4 + 1];
        c += partials[i * 4 + 2];
    }
    #pragma unroll
    for (int s = 16; s >= 1; s >>= 1) {
        a += __shfl_xor(a, s, 32);
        b += __shfl_xor(b, s, 32);
        c += __shfl_xor(c, s, 32);
    }
    if (lane == 0) { sred[0][wave] = a; sred[1][wave] = b; sred[2][wave] = c; }
    __syncthreads();
    if (tid == 0) {
        float pa = 0.0f, pb = 0.0f, pc = 0.0f;
        for (int w = 0; w < 8; ++w) { pa += sred[0][w]; pb += sred[1][w]; pc += sred[2][w]; }
        const float denom     = fmaxf(1.0f, pc);     // max(1, n_pos)
        const float objective = -(pa / denom) * 1.0f;  // PU_W = 1.0
        out[0] = objective;
        out[1] = objective;
        out[2] = pb / pc;                            // CE mean over positives (div by n_pos)
    }
}

extern "C" void kernel_launch(void* const* d_in, const int* in_sizes, int n_in,
                              void* d_out, int out_size, void* d_ws, size_t ws_size,
                              hipStream_t stream)
{
    const float* outputs   = (const float*)d_in[0];   // [65536, 1000] f32
    const int*   labels    = (const int*)  d_in[1];   // [65536]
    const float* priorlist = (const float*)d_in[2];   // [1000]
    // d_in[3] (indexlist) unused by the returned values of the reference.

    float* partials = (float*)d_ws;                   // NBLOCKS * 4 floats = 16 KB

    mpu_main<<<NBLOCKS, 256, 0, stream>>>(outputs, labels, priorlist, partials);
    mpu_finalize<<<1, 256, 0, stream>>>(partials, (float*)d_out);
}